// CN_4269197492713
// MI455X (gfx1250) — compile-verified
//
#include <hip/hip_runtime.h>

#define DEGREE           64
#define WAVES_PER_BLOCK  8
#define THREADS          (WAVES_PER_BLOCK * 32)

__global__ __launch_bounds__(THREADS) void cn_count_kernel(
    const int* __restrict__ adj,
    const int* __restrict__ nodes1,
    const int* __restrict__ nodes2,
    int* __restrict__ out,
    int n_pairs)
{
    // one 64-int sorted row of nbv per wave, wave-private
    __shared__ int sv[WAVES_PER_BLOCK][DEGREE];

    const int lane = threadIdx.x & 31;
    const int wave = threadIdx.x >> 5;
    const int p = blockIdx.x * WAVES_PER_BLOCK + wave;
    if (p >= n_pairs) return;   // wave-uniform exit

    // p is wave-uniform; make it provably uniform so the node-id fetches
    // become scalar loads (KMcnt) and row bases live in SGPRs.
    const int ps = __builtin_amdgcn_readfirstlane(p);
    const int u = nodes1[ps];
    const int v = nodes2[ps];
    const int* rowU = adj + (long long)u * DEGREE;
    const int* rowV = adj + (long long)v * DEGREE;

    // --- CDNA5 async copy: stage v's row into LDS (32 lanes x 8B = 256B) ---
    // GVS addressing: SGPR-pair base + per-lane 32-bit byte offset.
    {
        unsigned lds_addr = (unsigned)(unsigned long long)(&sv[wave][lane * 2]);
        unsigned voff = (unsigned)(lane * 8);
        asm volatile("global_load_async_to_lds_b64 %0, %1, %2"
                     :: "v"(lds_addr), "v"(voff), "s"(rowV)
                     : "memory");
    }

    // u's row straight into registers (two coalesced 128B loads per wave)
    int u0 = rowU[lane];
    int u1 = rowU[lane + 32];

    // first-occurrence flags within the sorted u-row (set semantics)
    int pu0   = __shfl_up(u0, 1, 32);
    int pu1   = __shfl_up(u1, 1, 32);
    int u0_31 = __shfl(u0, 31, 32);
    bool first0 = (lane == 0) ? true : (u0 != pu0);
    bool first1 = (lane == 0) ? (u1 != u0_31) : (u1 != pu1);

    // wait for this wave's async LDS fill (wave-private slice: no barrier)
    asm volatile("s_wait_asynccnt 0x0" ::: "memory");

    const int* S = &sv[wave][0];

    // branchless lower_bound membership test over S[0..63]
    // (row spans all 64 LDS banks once -> same-bank lane accesses are
    //  same-address broadcasts -> conflict-free)
    auto member = [&](int x) -> bool {
        int base = 0;
        if (S[base + 31] < x) base += 32;
        if (S[base + 15] < x) base += 16;
        if (S[base +  7] < x) base += 8;
        if (S[base +  3] < x) base += 4;
        if (S[base +  1] < x) base += 2;
        if (S[base     ] < x) base += 1;
        int w  = S[base];
        int lb = base + (w < x);          // lower_bound in [0,64]
        int cand = S[lb & (DEGREE - 1)];
        return (lb < DEGREE) && (cand == x);
    };

    bool hit0 = member(u0) && first0;
    bool hit1 = member(u1) && first1;

    unsigned long long b0 = __ballot(hit0);   // wave32: low 32 bits
    unsigned long long b1 = __ballot(hit1);
    if (lane == 0) {
        out[p] = (int)(__popcll(b0) + __popcll(b1));
    }
}

extern "C" void kernel_launch(void* const* d_in, const int* in_sizes, int n_in,
                              void* d_out, int out_size, void* d_ws, size_t ws_size,
                              hipStream_t stream) {
    (void)n_in; (void)d_ws; (void)ws_size;
    const int* adj    = (const int*)d_in[0];   // [N_NODES, DEGREE] int32, rows sorted
    const int* nodes1 = (const int*)d_in[1];   // [P] int32
    const int* nodes2 = (const int*)d_in[2];   // [P] int32
    int* out = (int*)d_out;                    // [P] int32

    const int P = in_sizes[1];
    const int grid = (P + WAVES_PER_BLOCK - 1) / WAVES_PER_BLOCK;
    cn_count_kernel<<<grid, THREADS, 0, stream>>>(adj, nodes1, nodes2, out, P);
}